// PromptPool_5669356830722
// MI455X (gfx1250) — compile-verified
//
#include <hip/hip_runtime.h>

#define B_    2048
#define POOL_ 1024
#define LEN_  8
#define DIM_  768
#define TOPK_ 5
#define VLEN_ (LEN_*DIM_)   // 6144 floats per codebook entry

typedef __attribute__((ext_vector_type(2))) float v2f;
typedef __attribute__((ext_vector_type(4))) float v4f;
typedef __attribute__((ext_vector_type(8))) float v8f;

// ---------------------------------------------------------------------------
// Kernel 0: row norms. One wave32 per row; waves [0,B) -> q_sq, [B,B+P) -> k_sq.
// ---------------------------------------------------------------------------
__global__ __launch_bounds__(256) void norms_kernel(const float* __restrict__ q,
                                                    const float* __restrict__ pk,
                                                    float* __restrict__ q_sq,
                                                    float* __restrict__ k_sq) {
    int wave = (blockIdx.x * blockDim.x + threadIdx.x) >> 5;
    int lane = threadIdx.x & 31;
    const float* src;
    float* dst;
    if (wave < B_) {
        src = q + (size_t)wave * DIM_;
        dst = q_sq + wave;
    } else {
        int row = wave - B_;
        if (row >= POOL_) return;
        src = pk + (size_t)row * DIM_;
        dst = k_sq + row;
    }
    float s = 0.0f;
    for (int i = lane; i < DIM_; i += 32) {
        float v = src[i];
        s = fmaf(v, v, s);
    }
    #pragma unroll
    for (int off = 16; off > 0; off >>= 1) s += __shfl_down(s, off, 32);
    if (lane == 0) *dst = s;
}

// ---------------------------------------------------------------------------
// Kernel 1: distance GEMM with V_WMMA_F32_16X16X4_F32.
// One wave per 16(M) x 64(N) strip: 4 f32 accumulators, K-loop over DIM in
// steps of 4, 4x A-reuse. All addressing is 32-bit per-lane offsets from the
// uniform kernarg bases so the backend emits global (not flat) loads.
// Epilogue fuses sqrt(max(q^2 + k^2 - 2*dot, 0)).
// ---------------------------------------------------------------------------
__global__ __launch_bounds__(32) void dist_kernel(const float* __restrict__ Q,
                                                  const float* __restrict__ Kp,
                                                  const float* __restrict__ q_sq,
                                                  const float* __restrict__ k_sq,
                                                  float* __restrict__ dist) {
    const int lane = threadIdx.x & 31;
    const int hi   = lane >> 4;     // 0 or 1
    const int lo   = lane & 15;
    const int mbase = blockIdx.y * 16;   // query-row tile
    const int nbase = blockIdx.x * 64;   // pool-col strip

    // A: lane holds Q[mbase+lo][k0 + 2*hi + {0,1}]        (16x4 f32 A layout)
    const int aoff = (mbase + lo) * DIM_ + 2 * hi;
    // B_j: lane holds Kp[nbase + j*16 + lo][k0 + 2*hi + {0,1}]  (B = keys^T)
    const int boff0 = (nbase + lo) * DIM_ + 2 * hi;
    const int boff1 = boff0 + 16 * DIM_;
    const int boff2 = boff0 + 32 * DIM_;
    const int boff3 = boff0 + 48 * DIM_;

    v8f acc0 = (v8f){0.f,0.f,0.f,0.f,0.f,0.f,0.f,0.f};
    v8f acc1 = acc0, acc2 = acc0, acc3 = acc0;

    for (int k0 = 0; k0 < DIM_; k0 += 4) {
        v2f a  = *(const v2f*)(Q  + aoff  + k0);
        v2f b0 = *(const v2f*)(Kp + boff0 + k0);
        v2f b1 = *(const v2f*)(Kp + boff1 + k0);
        v2f b2 = *(const v2f*)(Kp + boff2 + k0);
        v2f b3 = *(const v2f*)(Kp + boff3 + k0);
        acc0 = __builtin_amdgcn_wmma_f32_16x16x4_f32(false, a, false, b0, (short)0, acc0, false, false);
        acc1 = __builtin_amdgcn_wmma_f32_16x16x4_f32(false, a, false, b1, (short)0, acc1, false, false);
        acc2 = __builtin_amdgcn_wmma_f32_16x16x4_f32(false, a, false, b2, (short)0, acc2, false, false);
        acc3 = __builtin_amdgcn_wmma_f32_16x16x4_f32(false, a, false, b3, (short)0, acc3, false, false);
    }

    float ksq0 = k_sq[nbase + lo];
    float ksq1 = k_sq[nbase + 16 + lo];
    float ksq2 = k_sq[nbase + 32 + lo];
    float ksq3 = k_sq[nbase + 48 + lo];

    #pragma unroll
    for (int v = 0; v < 8; ++v) {
        int m = mbase + v + 8 * hi;        // C/D layout: VGPR v -> row v + 8*hi
        float qsm = q_sq[m];
        float* drow = dist + (size_t)m * POOL_ + nbase + lo;
        drow[ 0] = sqrtf(fmaxf(qsm + ksq0 - 2.0f * acc0[v], 0.0f));
        drow[16] = sqrtf(fmaxf(qsm + ksq1 - 2.0f * acc1[v], 0.0f));
        drow[32] = sqrtf(fmaxf(qsm + ksq2 - 2.0f * acc2[v], 0.0f));
        drow[48] = sqrtf(fmaxf(qsm + ksq3 - 2.0f * acc3[v], 0.0f));
    }
}

// ---------------------------------------------------------------------------
// Kernel 2: per-row top-5 smallest. One wave per row; per-lane insertion sort
// over 32 strided columns, then lane-0 LDS merge of 160 candidates.
// ---------------------------------------------------------------------------
__global__ __launch_bounds__(32) void topk_kernel(const float* __restrict__ dist,
                                                  int* __restrict__ top_idx,
                                                  float* __restrict__ rowsum) {
    __shared__ float sd[32 * TOPK_];
    __shared__ int   si[32 * TOPK_];
    const int b    = blockIdx.x;
    const int lane = threadIdx.x;
    const float* row = dist + (size_t)b * POOL_;

    float bd[TOPK_];
    int   bi[TOPK_];
    #pragma unroll
    for (int i = 0; i < TOPK_; ++i) { bd[i] = 3.4e38f; bi[i] = 0; }

    for (int p = lane; p < POOL_; p += 32) {
        float d = row[p];
        if (d < bd[TOPK_ - 1]) {
            int j = TOPK_ - 1;
            while (j > 0 && bd[j - 1] > d) {
                bd[j] = bd[j - 1]; bi[j] = bi[j - 1]; --j;
            }
            bd[j] = d; bi[j] = p;
        }
    }
    #pragma unroll
    for (int i = 0; i < TOPK_; ++i) {
        sd[lane * TOPK_ + i] = bd[i];
        si[lane * TOPK_ + i] = bi[i];
    }
    __syncthreads();
    if (lane == 0) {
        float fd[TOPK_];
        int   fi[TOPK_];
        #pragma unroll
        for (int i = 0; i < TOPK_; ++i) { fd[i] = 3.4e38f; fi[i] = 0; }
        for (int c = 0; c < 32 * TOPK_; ++c) {
            float d = sd[c];
            if (d < fd[TOPK_ - 1]) {
                int j = TOPK_ - 1;
                while (j > 0 && fd[j - 1] > d) {
                    fd[j] = fd[j - 1]; fi[j] = fi[j - 1]; --j;
                }
                fd[j] = d; fi[j] = si[c];
            }
        }
        float s = 0.0f;
        #pragma unroll
        for (int i = 0; i < TOPK_; ++i) {
            top_idx[b * TOPK_ + i] = fi[i];
            s += fd[i];
        }
        rowsum[b] = s;
    }
}

// ---------------------------------------------------------------------------
// Kernel 3: gather — the bandwidth-bound stage (~252 MB of streaming stores).
// One block per (b,k); 24 KB contiguous b128 copies. Loads stay RT (the 25 MB
// prompt_values pool is reused across blocks and L2-resident); stores are
// NON-TEMPORAL: the 252 MB output exceeds the 192 MB L2, is never re-read,
// and must not evict the hot pool/dist working set.
// ---------------------------------------------------------------------------
__global__ __launch_bounds__(256) void gather_kernel(const float* __restrict__ pv,
                                                     const int* __restrict__ top_idx,
                                                     float* __restrict__ out) {
    const int blk = blockIdx.x;              // b*TOPK_ + k
    const int idx = top_idx[blk];
    const v4f* __restrict__ src = (const v4f*)(pv  + (size_t)idx * VLEN_);
    v4f*       __restrict__ dst = (v4f*)      (out + (size_t)blk * VLEN_);
    #pragma unroll 2
    for (int i = threadIdx.x; i < VLEN_ / 4; i += 256) {
        v4f v = src[i];
        __builtin_nontemporal_store(v, &dst[i]);
    }
}

// ---------------------------------------------------------------------------
// Kernel 4: deterministic loss reduction (no float atomics).
// ---------------------------------------------------------------------------
__global__ __launch_bounds__(256) void loss_kernel(const float* __restrict__ rowsum,
                                                   float* __restrict__ out_loss) {
    __shared__ float s[256];
    float v = 0.0f;
    for (int i = threadIdx.x; i < B_; i += 256) v += rowsum[i];
    s[threadIdx.x] = v;
    __syncthreads();
    #pragma unroll
    for (int off = 128; off > 0; off >>= 1) {
        if (threadIdx.x < off) s[threadIdx.x] += s[threadIdx.x + off];
        __syncthreads();
    }
    if (threadIdx.x == 0) out_loss[0] = s[0] / (float)B_;
}

// ---------------------------------------------------------------------------
extern "C" void kernel_launch(void* const* d_in, const int* in_sizes, int n_in,
                              void* d_out, int out_size, void* d_ws, size_t ws_size,
                              hipStream_t stream) {
    const float* q  = (const float*)d_in[0];   // [B, D]
    const float* pk = (const float*)d_in[1];   // [P, D]
    const float* pv = (const float*)d_in[2];   // [P, L, D]
    // d_in[3] is top_k scalar (=5), baked into TOPK_.
    float* out = (float*)d_out;                // [B,K,L,D] floats + 1 loss

    // Workspace layout (all fully rewritten each call before being read):
    float* dist   = (float*)d_ws;                       // B*P floats = 8 MB
    float* q_sq   = dist + (size_t)B_ * POOL_;          // B
    float* k_sq   = q_sq + B_;                          // P
    float* rowsum = k_sq + POOL_;                       // B
    int*   tidx   = (int*)(rowsum + B_);                // B*TOPK_

    norms_kernel<<<(B_ + POOL_) / 8, 256, 0, stream>>>(q, pk, q_sq, k_sq);
    dist_kernel<<<dim3(POOL_ / 64, B_ / 16), 32, 0, stream>>>(q, pk, q_sq, k_sq, dist);
    topk_kernel<<<B_, 32, 0, stream>>>(dist, tidx, rowsum);
    gather_kernel<<<B_ * TOPK_, 256, 0, stream>>>(pv, tidx, out);
    loss_kernel<<<1, 256, 0, stream>>>(rowsum, out + (size_t)out_size - 1);
}